// RetinaNet_80083960201687
// MI455X (gfx1250) — compile-verified
//
#include <hip/hip_runtime.h>
#include <hip/hip_bf16.h>

#define A_TOTAL   76725
#define NCLS      80
#define TOPK      1000
#define TBPAD     1008      // 63 * 16 row tiles
#define LROWS     1024      // padded LDS box count (cols go to 1023)
#define SCORE_THR 0.001f
#define IOU_THR   0.5f

// ---- workspace layout (float elements) ----
#define OFF_BOXES 0                         // A_TOTAL float4  (306900 floats)
#define OFF_TOPV  306900                    // NCLS*TOPK
#define OFF_TB    386900                    // NCLS*TBPAD float4 (322560 floats, 16B aligned)
#define OFF_MASK  709460                    // NCLS*TOPK*32 uints

typedef float v2f __attribute__((ext_vector_type(2)));
typedef float v8f __attribute__((ext_vector_type(8)));

// ---------------------------------------------------------------------------
// 1) Decode + clip boxes (elementwise, bandwidth-trivial)
// ---------------------------------------------------------------------------
__global__ __launch_bounds__(256) void decode_kernel(
    const float4* __restrict__ anchors, const float4* __restrict__ deltas,
    float4* __restrict__ boxes) {
  int a = blockIdx.x * blockDim.x + threadIdx.x;
  if (a >= A_TOTAL) return;
  float4 an = anchors[a];
  float4 dl = deltas[a];
  float aw = an.z - an.x, ah = an.w - an.y;
  float cx = an.x + 0.5f * aw, cy = an.y + 0.5f * ah;
  float dx = dl.x * 0.1f, dy = dl.y * 0.1f;
  float dw = dl.z * 0.2f, dh = dl.w * 0.2f;
  float pcx = cx + dx * aw, pcy = cy + dy * ah;
  float pw = expf(dw) * aw, ph = expf(dh) * ah;
  float4 o;
  o.x = fmaxf(pcx - 0.5f * pw, 0.0f);
  o.y = fmaxf(pcy - 0.5f * ph, 0.0f);
  o.z = fminf(pcx + 0.5f * pw, 640.0f);
  o.w = fminf(pcy + 0.5f * ph, 640.0f);
  boxes[a] = o;
}

// ---------------------------------------------------------------------------
// 2) Per-class exact top-1000: 3-pass radix select on float bit patterns
//    (scores >= 0, so bits are order-isomorphic), then collect candidates
//    >= threshold and bitonic-sort 2048 (value desc, index asc tie-break).
// ---------------------------------------------------------------------------
__global__ __launch_bounds__(1024) void topk_kernel(
    const float* __restrict__ cls, const float4* __restrict__ boxes,
    float* __restrict__ topv, float4* __restrict__ tb) {
  __shared__ unsigned int hist[2048];
  __shared__ unsigned long long cand[2048];
  __shared__ unsigned int s_sel, s_rem, s_cnt;
  const int c = blockIdx.x;
  const int tid = threadIdx.x;
  const unsigned utid = (unsigned)tid;

  // ---- pass 1: top 11 bits ----
  for (int i = tid; i < 2048; i += 1024) hist[i] = 0;
  __syncthreads();
  for (int a = tid; a < A_TOTAL; a += 1024) {
    unsigned int bits = __float_as_uint(cls[a * NCLS + c]);
    atomicAdd(&hist[bits >> 21], 1u);
  }
  __syncthreads();
  if (tid == 0) {
    unsigned int cum = 0;
    for (int b = 2047; b >= 0; --b) {
      cum += hist[b];
      if (cum >= (unsigned)TOPK) { s_sel = (unsigned)b; s_rem = TOPK - (cum - hist[b]); break; }
    }
  }
  __syncthreads();
  unsigned int t1 = s_sel, rem = s_rem;
  __syncthreads();

  // ---- pass 2: middle 11 bits ----
  for (int i = tid; i < 2048; i += 1024) hist[i] = 0;
  __syncthreads();
  for (int a = tid; a < A_TOTAL; a += 1024) {
    unsigned int bits = __float_as_uint(cls[a * NCLS + c]);
    if ((bits >> 21) == t1) atomicAdd(&hist[(bits >> 10) & 0x7FFu], 1u);
  }
  __syncthreads();
  if (tid == 0) {
    unsigned int cum = 0;
    for (int b = 2047; b >= 0; --b) {
      cum += hist[b];
      if (cum >= rem) { s_sel = (unsigned)b; s_rem = rem - (cum - hist[b]); break; }
    }
  }
  __syncthreads();
  unsigned int t2 = s_sel; rem = s_rem;
  __syncthreads();

  // ---- pass 3: low 10 bits ----
  if (tid < 1024) hist[tid] = 0;
  __syncthreads();
  const unsigned int pfx = (t1 << 11) | t2;
  for (int a = tid; a < A_TOTAL; a += 1024) {
    unsigned int bits = __float_as_uint(cls[a * NCLS + c]);
    if ((bits >> 10) == pfx) atomicAdd(&hist[bits & 0x3FFu], 1u);
  }
  __syncthreads();
  if (tid == 0) {
    unsigned int cum = 0;
    for (int b = 1023; b >= 0; --b) {
      cum += hist[b];
      if (cum >= rem) { s_sel = (unsigned)b; break; }
    }
    s_cnt = 0;
  }
  __syncthreads();
  const unsigned int T = (t1 << 21) | (t2 << 10) | s_sel;

  // ---- collect candidates >= T (count >= K by construction) ----
  for (int a = tid; a < A_TOTAL; a += 1024) {
    unsigned int bits = __float_as_uint(cls[a * NCLS + c]);
    if (bits >= T) {
      unsigned int pos = atomicAdd(&s_cnt, 1u);
      if (pos < 2048u)
        cand[pos] = ((unsigned long long)bits << 32) |
                    (unsigned long long)(0xFFFFFFFFu - (unsigned)a);
    }
  }
  __syncthreads();
  unsigned int cnt = s_cnt; if (cnt > 2048u) cnt = 2048u;
  for (int i = tid; i < 2048; i += 1024)
    if ((unsigned)i >= cnt) cand[i] = 0ull;
  __syncthreads();

  // ---- bitonic sort, descending, N=2048 with 1024 threads ----
  for (unsigned size = 2; size <= 2048u; size <<= 1) {
    for (unsigned stride = size >> 1; stride > 0; stride >>= 1) {
      __syncthreads();
      unsigned pos = 2u * utid - (utid & (stride - 1u));
      bool desc = ((pos & size) == 0u);
      unsigned long long x = cand[pos], y = cand[pos + stride];
      if ((x < y) == desc) { cand[pos] = y; cand[pos + stride] = x; }
    }
  }
  __syncthreads();

  // ---- emit top-K values + gathered boxes (pad rows 1000..1007 with 0) ----
  if (tid < TOPK) {
    unsigned long long key = cand[tid];
    if (key != 0ull) {
      float v = __uint_as_float((unsigned int)(key >> 32));
      unsigned int idx = 0xFFFFFFFFu - (unsigned int)(key & 0xFFFFFFFFull);
      topv[c * TOPK + tid] = v;
      tb[c * TBPAD + tid] = boxes[idx];
    } else {
      topv[c * TOPK + tid] = 0.0f;
      tb[c * TBPAD + tid] = make_float4(0.f, 0.f, 0.f, 0.f);
    }
  } else if (tid < TBPAD) {
    tb[c * TBPAD + tid] = make_float4(0.f, 0.f, 0.f, 0.f);
  }
}

// ---------------------------------------------------------------------------
// 3) Suppression-mask builder. Each wave owns one (16-row-tile, 32-col word).
//    Boxes are staged memory->LDS with GLOBAL_LOAD_ASYNC_TO_LDS_B128 (CDNA5
//    async copy, ASYNCcnt-tracked, no VGPR round-trip). Pairwise denominator
//    area_i + area_j is a rank-2 matmul on V_WMMA_F32_16X16X4_F32:
//    A=[area_i|1] (16x4), B=[1 ; area_j] (4x16). The iou>thr test is done as
//    inter > thr*union (union>0) to avoid IEEE division chains. ballot_w32
//    packs the 16x16 predicate tile: low16 -> row v, high16 -> row v+8; two
//    col-tiles assemble one 32-bit mask word (plain store, each (row,word)
//    written exactly once per launch).
// ---------------------------------------------------------------------------
__global__ __launch_bounds__(256) void iou_mask_kernel(
    const float4* __restrict__ tb, unsigned int* __restrict__ mask) {
  __shared__ float4 lbox[LROWS];
  __shared__ float  larea[LROWS];
  const int c = blockIdx.x;
  const int tid = threadIdx.x;

  // --- async global->LDS staging of this class's 1008 candidate boxes ---
  {
    // low 32 bits of a generic pointer to LDS == hardware LDS byte address
    unsigned lds_base = (unsigned)(unsigned long long)(&lbox[0]);
    for (int i = tid; i < TBPAD; i += 256) {
      unsigned ldst = lds_base + (unsigned)i * 16u;
      const float4* gsrc = &tb[c * TBPAD + i];
      asm volatile("global_load_async_to_lds_b128 %0, %1, off"
                   :: "v"(ldst), "v"(gsrc) : "memory");
    }
    for (int i = TBPAD + tid; i < LROWS; i += 256)
      lbox[i] = make_float4(0.f, 0.f, 0.f, 0.f);
    asm volatile("s_wait_asynccnt 0x0" ::: "memory");
  }
  __syncthreads();
  for (int i = tid; i < LROWS; i += 256) {
    float4 b = lbox[i];
    larea[i] = (b.z - b.x) * (b.w - b.y);
  }
  __syncthreads();

  const int wave = tid >> 5;
  const int lane = tid & 31;
  const int task = blockIdx.y * 8 + wave;   // 0..2015 == 63 row-tiles * 32 words
  const int ti = task >> 5;                 // row tile 0..62
  const int w  = task & 31;                 // 32-col mask word
  const int i0 = ti * 16;
  const int hi = lane >> 4;                 // 0: lanes 0-15, 1: lanes 16-31
  const int n16 = lane & 15;

  // A matrix 16x4: lanes 0-15 hold (K0,K1)=(area_i[M],1); lanes 16-31 hold (K2,K3)=0
  v2f av;
  av.x = hi ? 0.0f : larea[i0 + n16];
  av.y = hi ? 0.0f : 1.0f;

  // per-lane row boxes (jt-invariant): D VGPR v -> row v (+8 for hi lanes)
  float4 bi[8];
#pragma unroll
  for (int v = 0; v < 8; ++v) bi[v] = lbox[i0 + v + 8 * hi];

  unsigned int acc = 0;
#pragma unroll
  for (int jt = 0; jt < 2; ++jt) {
    const int j0 = w * 32 + jt * 16;
    // B matrix 4x16: VGPR0 lanes 0-15 = row K0 (ones), lanes 16-31 = row K1 (area_j)
    v2f bv;
    bv.x = hi ? larea[j0 + n16] : 1.0f;
    bv.y = 0.0f;
    v8f c8 = {0.f, 0.f, 0.f, 0.f, 0.f, 0.f, 0.f, 0.f};
    v8f d;
#if __has_builtin(__builtin_amdgcn_wmma_f32_16x16x4_f32)
    d = __builtin_amdgcn_wmma_f32_16x16x4_f32(false, av, false, bv,
                                              (short)0, c8, false, false);
#else
#pragma unroll
    for (int v = 0; v < 8; ++v) d[v] = larea[i0 + v + 8 * hi] + larea[j0 + n16];
    (void)c8;
#endif
    const int gj = j0 + n16;
    float4 bj = lbox[gj];
    unsigned int balSel = 0;
#pragma unroll
    for (int v = 0; v < 8; ++v) {
      const int gi = i0 + v + 8 * hi;
      float xx1 = fmaxf(bi[v].x, bj.x);
      float yy1 = fmaxf(bi[v].y, bj.y);
      float xx2 = fminf(bi[v].z, bj.z);
      float yy2 = fminf(bi[v].w, bj.w);
      float inter = fmaxf(xx2 - xx1, 0.0f) * fmaxf(yy2 - yy1, 0.0f);
      // iou > thr  <=>  inter > thr * (union)   (union = d[v]-inter+eps > 0)
      bool pred = (inter > IOU_THR * (d[v] - inter + 1e-8f)) &&
                  (gj > gi) && (gi < TOPK) && (gj < TOPK);
      unsigned int bal = (unsigned int)__builtin_amdgcn_ballot_w32(pred);
      if ((lane & 7) == v) balSel = bal;    // lane L<16 keeps ballot for v==L&7
    }
    if (lane < 16) {
      unsigned int half = (lane >> 3) ? (balSel >> 16) : (balSel & 0xFFFFu);
      acc |= half << (16 * jt);
    }
  }
  if (lane < 16) {
    int row = i0 + (lane & 7) + 8 * (lane >> 3);
    if (row < TOPK) mask[c * TOPK * 32 + row * 32 + w] = acc;
  }
}

// ---------------------------------------------------------------------------
// 4) Sequential NMS reduction (wave32: lane w owns keep word w, shfl
//    broadcasts bit i) + padded output emission.
// ---------------------------------------------------------------------------
__global__ __launch_bounds__(1024) void nms_out_kernel(
    const float* __restrict__ topv, const float4* __restrict__ tb,
    const unsigned int* __restrict__ mask, float* __restrict__ out) {
  __shared__ unsigned int kw[32];
  const int c = blockIdx.x;
  const int tid = threadIdx.x;
  if (tid < 32) {
    unsigned int keepw = 0;
    for (int b = 0; b < 32; ++b) {
      int k = tid * 32 + b;
      if (k < TOPK && topv[c * TOPK + k] > SCORE_THR) keepw |= (1u << b);
    }
    const unsigned int* mrow = mask + c * TOPK * 32;
    for (int i = 0; i < TOPK; ++i) {
      unsigned int cur = __shfl(keepw, i >> 5, 32);
      if ((cur >> (i & 31)) & 1u) keepw &= ~mrow[i * 32 + tid];
    }
    kw[tid] = keepw;
  }
  __syncthreads();
  for (int k = tid; k < TOPK; k += 1024) {
    int o = c * TOPK + k;
    bool kp = (kw[k >> 5] >> (k & 31)) & 1u;
    float v = topv[o];
    float4 b = tb[c * TBPAD + k];
    out[o] = kp ? v : 0.0f;                               // scores * keep
    out[NCLS * TOPK + o] = (float)c;                      // labels
    float4 ob = kp ? b : make_float4(0.f, 0.f, 0.f, 0.f); // boxes * keep
    *reinterpret_cast<float4*>(out + 2 * NCLS * TOPK + (size_t)o * 4) = ob;
    out[6 * NCLS * TOPK + o] = kp ? 1.0f : 0.0f;          // keep flags
  }
}

// ---------------------------------------------------------------------------
extern "C" void kernel_launch(void* const* d_in, const int* in_sizes, int n_in,
                              void* d_out, int out_size, void* d_ws, size_t ws_size,
                              hipStream_t stream) {
  // inputs: [0]=img (unused), [1]=cls [1,A,80], [2]=reg [1,A,4], [3]=anchors [1,A,4]
  const float*  cls = (const float*)d_in[1];
  const float4* reg = (const float4*)d_in[2];
  const float4* anc = (const float4*)d_in[3];
  float* ws = (float*)d_ws;
  float4*       boxes = (float4*)(ws + OFF_BOXES);
  float*        topv  = ws + OFF_TOPV;
  float4*       tb    = (float4*)(ws + OFF_TB);
  unsigned int* mask  = (unsigned int*)(ws + OFF_MASK);
  float* out = (float*)d_out;

  decode_kernel<<<(A_TOTAL + 255) / 256, 256, 0, stream>>>(anc, reg, boxes);
  topk_kernel<<<NCLS, 1024, 0, stream>>>(cls, boxes, topv, tb);
  iou_mask_kernel<<<dim3(NCLS, 252), 256, 0, stream>>>(tb, mask);
  nms_out_kernel<<<NCLS, 1024, 0, stream>>>(topv, tb, mask, out);
}